// VectorQuantizerEMA_23313082483079
// MI455X (gfx1250) — compile-verified
//
#include <hip/hip_runtime.h>
#include <hip/hip_bf16.h>

// ---------------- problem constants ----------------
#define N_TOKENS   131072
#define K_CODES    1024
#define DIM        64
#define DECAY_F    0.99f
#define EPSILON_F  1e-5f
#define COMMIT_F   0.25f

typedef float v2f __attribute__((ext_vector_type(2)));
typedef float v4f __attribute__((ext_vector_type(4)));
typedef float v8f __attribute__((ext_vector_type(8)));

// ---------------------------------------------------------------------------
// Kernel 1: per-code squared norms + zero scratch accumulators / output scalars
// ---------------------------------------------------------------------------
__global__ void vq_prep(const float* __restrict__ embed,
                        float* __restrict__ e_norms,
                        float* __restrict__ counts,
                        float* __restrict__ dw,
                        float* __restrict__ loss_out,
                        float* __restrict__ perp_out) {
    int k = blockIdx.x * blockDim.x + threadIdx.x;
    if (k >= K_CODES) return;
    const v4f* row = (const v4f*)(embed + (size_t)k * DIM);
    float s = 0.0f;
#pragma unroll
    for (int i = 0; i < DIM / 4; ++i) {
        v4f v = row[i];
        s += v.x * v.x + v.y * v.y + v.z * v.z + v.w * v.w;
    }
    e_norms[k] = s;
    counts[k]  = 0.0f;
    v4f* dwr = (v4f*)(dw + (size_t)k * DIM);
#pragma unroll
    for (int i = 0; i < DIM / 4; ++i) dwr[i] = (v4f){0.f, 0.f, 0.f, 0.f};
    if (k == 0) { *loss_out = 0.0f; *perp_out = 0.0f; }
}

// ---------------------------------------------------------------------------
// Kernel 2: WMMA fp32 distance GEMM + argmin.
// One wave32 per 16-token tile. For each chunk of 16 codes, accumulate
// dot(x, e) over D=64 with 16x v_wmma_f32_16x16x4_f32 split across TWO
// accumulators (even/odd K-step) so adjacent WMMAs are independent and can
// dual-issue instead of RAW-chaining through one D register set.
// C/D layout: VGPR r -> row (r + 8*(lane>>4)), column (lane & 15).
// A 16x4 frag: lanes 0-15 hold K={4s,4s+1}, lanes 16-31 hold K={4s+2,4s+3}.
// ---------------------------------------------------------------------------
__launch_bounds__(256)
__global__ void vq_argmin_wmma(const float* __restrict__ x,
                               const float* __restrict__ embed,
                               const float* __restrict__ e_norms,
                               int* __restrict__ indices) {
    const int lane  = threadIdx.x & 31;
    const int wave  = threadIdx.x >> 5;
    const int tile  = blockIdx.x * 8 + wave;
    const int tbase = tile * 16;
    const int row   = lane & 15;   // A row / B column within the tile
    const int half  = lane >> 4;   // K sub-offset selector

    // Preload A fragments: the whole 16x64 input tile lives in 16 v2f regs/lane.
    v2f a[16];
    const float* arow = x + (size_t)(tbase + row) * DIM + 2 * half;
#pragma unroll
    for (int s = 0; s < 16; ++s) a[s] = *(const v2f*)(arow + 4 * s);

    float bestv[8];
    int   besti[8];
#pragma unroll
    for (int r = 0; r < 8; ++r) { bestv[r] = 3.4e38f; besti[r] = 0; }

    for (int c = 0; c < K_CODES / 16; ++c) {
        const int cbase = c * 16;
        const float* brow = embed + (size_t)(cbase + row) * DIM + 2 * half;
        v8f acc0 = (v8f){0.f, 0.f, 0.f, 0.f, 0.f, 0.f, 0.f, 0.f};
        v8f acc1 = (v8f){0.f, 0.f, 0.f, 0.f, 0.f, 0.f, 0.f, 0.f};
#pragma unroll
        for (int s = 0; s < 16; s += 2) {
            v2f b0 = *(const v2f*)(brow + 4 * s);
            v2f b1 = *(const v2f*)(brow + 4 * (s + 1));
            acc0 = __builtin_amdgcn_wmma_f32_16x16x4_f32(
                false, a[s],     false, b0, (short)0, acc0, false, false);
            acc1 = __builtin_amdgcn_wmma_f32_16x16x4_f32(
                false, a[s + 1], false, b1, (short)0, acc1, false, false);
        }
        const float en  = e_norms[cbase + row];  // this lane's column norm
        const int   col = cbase + row;
#pragma unroll
        for (int r = 0; r < 8; ++r) {
            float score = en - 2.0f * (acc0[r] + acc1[r]);
            if (score < bestv[r]) { bestv[r] = score; besti[r] = col; }
        }
    }

    // Min-reduce across the 16 lanes holding the same rows (masks stay in-half).
#pragma unroll
    for (int r = 0; r < 8; ++r) {
#pragma unroll
        for (int m = 1; m < 16; m <<= 1) {
            float ov = __shfl_xor(bestv[r], m, 32);
            int   oi = __shfl_xor(besti[r], m, 32);
            if (ov < bestv[r] || (ov == bestv[r] && oi < besti[r])) {
                bestv[r] = ov; besti[r] = oi;
            }
        }
    }
    if (row == 0) {
#pragma unroll
        for (int r = 0; r < 8; ++r) indices[tbase + half * 8 + r] = besti[r];
    }
}

// ---------------------------------------------------------------------------
// Kernel 3: one-hot encodings (537 MB write -> the bandwidth floor).
// Block per token; non-temporal 8-byte vector stores (slab base is only
// 8B-aligned, and this stream is never re-read on device -> keep it out of L2).
// ---------------------------------------------------------------------------
__global__ void vq_encodings(const int* __restrict__ indices,
                             float* __restrict__ enc) {
    const int n   = blockIdx.x;
    const int t   = threadIdx.x;
    const int idx = indices[n];
    float* rowp = enc + (size_t)n * K_CODES;
#pragma unroll
    for (int i = 0; i < 2; ++i) {
        int c = i * 512 + 2 * t;
        v2f v = (v2f){0.f, 0.f};
        if (idx == c)     v.x = 1.0f;
        if (idx == c + 1) v.y = 1.0f;
        __builtin_nontemporal_store(v, (v2f*)(rowp + c));
    }
}

// ---------------------------------------------------------------------------
// Kernel 4: segment-sum dw and cluster counts (fp32 global atomics).
// Wave per token; lane covers dims {lane, lane+32}.
// ---------------------------------------------------------------------------
__global__ void vq_stats(const float* __restrict__ x,
                         const int* __restrict__ indices,
                         float* __restrict__ counts,
                         float* __restrict__ dw) {
    const int lane = threadIdx.x & 31;
    const int wave = threadIdx.x >> 5;
    const int n    = blockIdx.x * 8 + wave;
    const int idx  = indices[n];
    float x0 = x[(size_t)n * DIM + lane];
    float x1 = x[(size_t)n * DIM + lane + 32];
    atomicAdd(dw + (size_t)idx * DIM + lane,      x0);
    atomicAdd(dw + (size_t)idx * DIM + lane + 32, x1);
    if (lane == 0) atomicAdd(counts + idx, 1.0f);
}

// ---------------------------------------------------------------------------
// Kernel 5: EMA update + Laplace smoothing + embedding_new + perplexity.
// One block, K=1024 threads, LDS tree reductions.
// ---------------------------------------------------------------------------
__launch_bounds__(1024)
__global__ void vq_update(const float* __restrict__ ema_w,
                          const float* __restrict__ ema_cs,
                          const float* __restrict__ counts,
                          const float* __restrict__ dw,
                          float* __restrict__ emb_new,
                          float* __restrict__ perp_out) {
    __shared__ float red[K_CODES];
    const int k = threadIdx.x;
    const float cnt = counts[k];
    float cs = ema_cs[k] * DECAY_F + (1.0f - DECAY_F) * cnt;

    red[k] = cs;
    __syncthreads();
    for (int off = 512; off > 0; off >>= 1) {
        if (k < off) red[k] += red[k + off];
        __syncthreads();
    }
    const float nsum = red[0];
    __syncthreads();

    const float p = cnt * (1.0f / (float)N_TOKENS);
    red[k] = p * __logf(p + 1e-10f);
    __syncthreads();
    for (int off = 512; off > 0; off >>= 1) {
        if (k < off) red[k] += red[k + off];
        __syncthreads();
    }
    if (k == 0) *perp_out = __expf(-red[0]);

    const float csm = (cs + EPSILON_F) / (nsum + (float)K_CODES * EPSILON_F) * nsum;
    const float inv = 1.0f / csm;
#pragma unroll
    for (int d = 0; d < DIM; ++d) {
        emb_new[(size_t)k * DIM + d] =
            (ema_w[(size_t)k * DIM + d] * DECAY_F +
             (1.0f - DECAY_F) * dw[(size_t)k * DIM + d]) * inv;
    }
}

// ---------------------------------------------------------------------------
// Kernel 6: quantized = embedding_new[idx]; commitment loss accumulation.
// Wave per token; non-temporal scalar stores (qout base is only 4B-aligned,
// and quantized is write-once/never re-read on device).
// ---------------------------------------------------------------------------
__global__ void vq_quantize(const float* __restrict__ x,
                            const int* __restrict__ indices,
                            const float* __restrict__ emb_new,
                            float* __restrict__ qout,
                            float* __restrict__ loss_out) {
    const int lane = threadIdx.x & 31;
    const int wave = threadIdx.x >> 5;
    const int n    = blockIdx.x * 8 + wave;
    const int idx  = indices[n];
    float x0 = x[(size_t)n * DIM + lane];
    float x1 = x[(size_t)n * DIM + lane + 32];
    float q0 = emb_new[(size_t)idx * DIM + lane];
    float q1 = emb_new[(size_t)idx * DIM + lane + 32];
    __builtin_nontemporal_store(q0, qout + (size_t)n * DIM + lane);
    __builtin_nontemporal_store(q1, qout + (size_t)n * DIM + lane + 32);
    float d0 = q0 - x0, d1 = q1 - x1;
    float s = d0 * d0 + d1 * d1;
#pragma unroll
    for (int m = 16; m > 0; m >>= 1) s += __shfl_xor(s, m, 32);
    if (lane == 0)
        atomicAdd(loss_out, s * (0.5f * COMMIT_F / (float)N_TOKENS));
}

// ---------------------------------------------------------------------------
extern "C" void kernel_launch(void* const* d_in, const int* in_sizes, int n_in,
                              void* d_out, int out_size, void* d_ws, size_t ws_size,
                              hipStream_t stream) {
    const float* x      = (const float*)d_in[0];  // [N, 64]
    const float* embed  = (const float*)d_in[1];  // [K, 64]
    const float* ema_w  = (const float*)d_in[2];  // [K, 64]
    const float* ema_cs = (const float*)d_in[3];  // [K]

    float* out      = (float*)d_out;
    float* loss_out = out;                                  // [0]
    float* qout     = out + 1;                              // [1, 1+N*D)
    float* perp_out = out + 1 + (size_t)N_TOKENS * DIM;     // scalar
    float* enc      = perp_out + 1;                         // [.., +N*K)

    float* e_norms = (float*)d_ws;                          // K
    float* counts  = e_norms + K_CODES;                     // K
    float* dw      = counts + K_CODES;                      // K*D
    float* emb_new = dw + (size_t)K_CODES * DIM;            // K*D
    int*   indices = (int*)(emb_new + (size_t)K_CODES * DIM); // N

    vq_prep       <<<K_CODES / 256, 256, 0, stream>>>(embed, e_norms, counts, dw,
                                                      loss_out, perp_out);
    vq_argmin_wmma<<<N_TOKENS / (16 * 8), 256, 0, stream>>>(x, embed, e_norms, indices);
    vq_encodings  <<<N_TOKENS, 256, 0, stream>>>(indices, enc);
    vq_stats      <<<N_TOKENS / 8, 256, 0, stream>>>(x, indices, counts, dw);
    vq_update     <<<1, 1024, 0, stream>>>(ema_w, ema_cs, counts, dw, emb_new, perp_out);
    vq_quantize   <<<N_TOKENS / 8, 256, 0, stream>>>(x, indices, emb_new, qout, loss_out);
}